// GeodesicIntegrator_5360119185651
// MI455X (gfx1250) — compile-verified
//
#include <hip/hip_runtime.h>

// Geodesic RK4 integrator for MI455X (gfx1250).
//
// Γ^i_jk = δ_ij g_k + δ_ik g_j − δ_jk g_i  =>  a_i = g_i|v|^2 − 2 v_i (g·v)
// so no Christoffel tensor is ever materialized: each accel eval needs only
// the 6-neighbor gradient of Phi at one clamped voxel.
//
// Workload is a serial latency chain (49 steps × 4 RK stages, B=2), so:
//  - block 0, lanes 0..1: one particle per lane, scalar integration.
//  - blocks 1..8: stream global_prefetch_b8 over all of Phi (7 MB << 192 MB L2)
//    concurrently, so the integrator's dependent loads hit L2, not HBM.

namespace {

constexpr int   kB      = 2;
constexpr int   kH      = 96;
constexpr int   kW      = 96;
constexpr int   kD      = 96;
constexpr int   kSteps  = 50;                    // MAX_STEPS
constexpr float kStep   = 0.1f;                  // STEP
constexpr float kMaxC   = 100.0f;                // MAX_CURV

__device__ __forceinline__ int clampi(int v, int lo, int hi) {
  return v < lo ? lo : (v > hi ? hi : v);
}

__device__ __forceinline__ float clampf(float v, float lo, float hi) {
  return fminf(fmaxf(v, lo), hi);
}

// a = clip(g*|v|^2 - 2*v*(g.v), +-100) with g = edge-padded central gradient.
__device__ __forceinline__ void accel(const float* __restrict__ phi_b,
                                      float px, float py, float pz,
                                      float vx, float vy, float vz,
                                      float& ax, float& ay, float& az) {
  // idx = clip(trunc(pos * 95), 0, 95)   (v_cvt_i32_f32 truncates like astype)
  const int ix = clampi((int)(px * 95.0f), 0, 95);
  const int iy = clampi((int)(py * 95.0f), 0, 95);
  const int iz = clampi((int)(pz * 95.0f), 0, 95);
  const int xm = ix > 0 ? ix - 1 : 0, xp = ix < 95 ? ix + 1 : 95;
  const int ym = iy > 0 ? iy - 1 : 0, yp = iy < 95 ? iy + 1 : 95;
  const int zm = iz > 0 ? iz - 1 : 0, zp = iz < 95 ? iz + 1 : 95;

  // Six independent loads: issued back-to-back, overlapped in the memory
  // pipeline, one L2 round-trip of latency per accel evaluation.
  const float fxp = phi_b[((size_t)xp * kW + iy) * kD + iz];
  const float fxm = phi_b[((size_t)xm * kW + iy) * kD + iz];
  const float fyp = phi_b[((size_t)ix * kW + yp) * kD + iz];
  const float fym = phi_b[((size_t)ix * kW + ym) * kD + iz];
  const float fzp = phi_b[((size_t)ix * kW + iy) * kD + zp];
  const float fzm = phi_b[((size_t)ix * kW + iy) * kD + zm];

  const float gx = (fxp - fxm) * 0.5f;   // /(2*DX), DX=1
  const float gy = (fyp - fym) * 0.5f;
  const float gz = (fzp - fzm) * 0.5f;

  const float gv = gx * vx + gy * vy + gz * vz;
  const float vv = vx * vx + vy * vy + vz * vz;

  ax = clampf(gx * vv - 2.0f * vx * gv, -kMaxC, kMaxC);
  ay = clampf(gy * vv - 2.0f * vy * gv, -kMaxC, kMaxC);
  az = clampf(gz * vv - 2.0f * vz * gv, -kMaxC, kMaxC);
}

}  // namespace

__global__ void __launch_bounds__(256)
geodesic_rk4_kernel(const float* __restrict__ start_pos,
                    const float* __restrict__ start_vel,
                    const float* __restrict__ Phi,
                    float* __restrict__ out) {
  if (blockIdx.x != 0) {
    // ---- L2 warmers: stream global_prefetch_b8 over all of Phi ----
    const char*  p      = (const char*)Phi;
    const size_t bytes  = (size_t)kB * kH * kW * kD * sizeof(float);
    const size_t line   = 128;
    const size_t nlines = bytes / line;
    const size_t tid  = (size_t)(blockIdx.x - 1) * blockDim.x + threadIdx.x;
    const size_t nthr = (size_t)(gridDim.x - 1) * blockDim.x;
    for (size_t i = tid; i < nlines; i += nthr) {
      __builtin_prefetch(p + i * line, /*rw=*/0, /*locality=*/1);
    }
    return;
  }

  // ---- Integrator: one particle per lane (wave32; lanes 0..1 active) ----
  const int b = threadIdx.x;
  if (b >= kB) return;

  const float* phi_b   = Phi + (size_t)b * kH * kW * kD;
  float*       pos_out = out + (size_t)b * kSteps * 3;
  float*       vel_out = out + (size_t)kB * kSteps * 3 + (size_t)b * kSteps * 3;

  float px = start_pos[b * 3 + 0];
  float py = start_pos[b * 3 + 1];
  float pz = start_pos[b * 3 + 2];
  float vx = start_vel[b * 3 + 0];
  float vy = start_vel[b * 3 + 1];
  float vz = start_vel[b * 3 + 2];

  pos_out[0] = px; pos_out[1] = py; pos_out[2] = pz;
  vel_out[0] = vx; vel_out[1] = vy; vel_out[2] = vz;

  const float h  = kStep;
  const float hh = 0.5f * kStep;
  const float h6 = (float)(0.1 / 6.0);  // match JAX's double->f32 narrowing

  for (int s = 1; s < kSteps; ++s) {
    // k1: k1v = v, k1a = a(pos, v)
    float k1ax, k1ay, k1az;
    accel(phi_b, px, py, pz, vx, vy, vz, k1ax, k1ay, k1az);

    // k2: k2v = v + hh*k1a, k2a = a(pos + hh*v, k2v)
    const float k2vx = vx + hh * k1ax;
    const float k2vy = vy + hh * k1ay;
    const float k2vz = vz + hh * k1az;
    float k2ax, k2ay, k2az;
    accel(phi_b, px + hh * vx, py + hh * vy, pz + hh * vz,
          k2vx, k2vy, k2vz, k2ax, k2ay, k2az);

    // k3: k3v = v + hh*k2a, k3a = a(pos + hh*k2v, k3v)
    const float k3vx = vx + hh * k2ax;
    const float k3vy = vy + hh * k2ay;
    const float k3vz = vz + hh * k2az;
    float k3ax, k3ay, k3az;
    accel(phi_b, px + hh * k2vx, py + hh * k2vy, pz + hh * k2vz,
          k3vx, k3vy, k3vz, k3ax, k3ay, k3az);

    // k4: k4v = v + h*k3a, k4a = a(pos + h*k3v, k4v)
    const float k4vx = vx + h * k3ax;
    const float k4vy = vy + h * k3ay;
    const float k4vz = vz + h * k3az;
    float k4ax, k4ay, k4az;
    accel(phi_b, px + h * k3vx, py + h * k3vy, pz + h * k3vz,
          k4vx, k4vy, k4vz, k4ax, k4ay, k4az);

    // pos += h/6 * (k1v + 2 k2v + 2 k3v + k4v)   (k1v = old v)
    px += h6 * (vx + 2.0f * k2vx + 2.0f * k3vx + k4vx);
    py += h6 * (vy + 2.0f * k2vy + 2.0f * k3vy + k4vy);
    pz += h6 * (vz + 2.0f * k2vz + 2.0f * k3vz + k4vz);
    // vel += h/6 * (k1a + 2 k2a + 2 k3a + k4a)
    vx += h6 * (k1ax + 2.0f * k2ax + 2.0f * k3ax + k4ax);
    vy += h6 * (k1ay + 2.0f * k2ay + 2.0f * k3ay + k4ay);
    vz += h6 * (k1az + 2.0f * k2az + 2.0f * k3az + k4az);

    pos_out[s * 3 + 0] = px;
    pos_out[s * 3 + 1] = py;
    pos_out[s * 3 + 2] = pz;
    vel_out[s * 3 + 0] = vx;
    vel_out[s * 3 + 1] = vy;
    vel_out[s * 3 + 2] = vz;
  }
}

extern "C" void kernel_launch(void* const* d_in, const int* in_sizes, int n_in,
                              void* d_out, int out_size, void* d_ws, size_t ws_size,
                              hipStream_t stream) {
  (void)in_sizes; (void)n_in; (void)out_size; (void)d_ws; (void)ws_size;
  const float* start_pos = (const float*)d_in[0];  // (B,3) f32
  const float* start_vel = (const float*)d_in[1];  // (B,3) f32
  const float* Phi       = (const float*)d_in[2];  // (B,96,96,96) f32
  float*       out       = (float*)d_out;          // 300 pos + 300 vel f32

  // Block 0 integrates; blocks 1..8 (2048 lanes) prefetch Phi (7 MB) into L2
  // concurrently — ~27 cachelines per lane, done in O(1 us).
  geodesic_rk4_kernel<<<9, 256, 0, stream>>>(start_pos, start_vel, Phi, out);
}